// approximator_33122787787496
// MI455X (gfx1250) — compile-verified
//
#include <hip/hip_runtime.h>
#include <stdint.h>

typedef float v2f __attribute__((ext_vector_type(2)));
typedef float v8f __attribute__((ext_vector_type(8)));

#define NSLICE (32 * 512)      // N*C independent 56x56 slices
#define HW 56
#define BLK 7
#define SLICE_ELEMS (HW * HW)  // 3136 floats
#define SLICE_F4 (SLICE_ELEMS / 4)  // 784 float4 chunks
#define WAVES_PER_BLOCK 2
#define PQ_OFF SLICE_ELEMS               // P|Q staging: 56 rows x 16 cols
#define D_OFF (SLICE_ELEMS + HW * 16)    // WMMA D staging: 8 x 16
#define WAVE_LDS (SLICE_ELEMS + HW * 16 + 128)  // 4160 floats per wave

__global__ __launch_bounds__(WAVES_PER_BLOCK * 32)
void approximator_blockmse_kernel(const float* __restrict__ x,
                                  float* __restrict__ out) {
  __shared__ __align__(16) float lds[WAVES_PER_BLOCK * WAVE_LDS];

  const int tid  = threadIdx.x;
  const int wave = tid >> 5;
  const int lane = tid & 31;
  const int s    = blockIdx.x * WAVES_PER_BLOCK + wave;  // slice id

  float* wl = lds + wave * WAVE_LDS;
  const uint32_t lds_base = (uint32_t)(uintptr_t)wl;  // low 32 bits = LDS offset

  // Wave-uniform 64-bit global base of this slice, forced into SGPRs.
  uint64_t sp = (uint64_t)(uintptr_t)(x + (size_t)s * SLICE_ELEMS);
  uint32_t sp_lo = (uint32_t)__builtin_amdgcn_readfirstlane((int)(uint32_t)sp);
  uint32_t sp_hi = (uint32_t)__builtin_amdgcn_readfirstlane((int)(uint32_t)(sp >> 32));
  uint64_t sbase = ((uint64_t)sp_hi << 32) | sp_lo;

  // ---- Stage 1: async-copy the whole slice global -> LDS (B128 per lane) ----
  for (int j = lane; j < SLICE_F4; j += 32) {
    uint32_t ldsoff = lds_base + (uint32_t)(j * 16);
    uint32_t voff   = (uint32_t)(j * 16);
    asm volatile("global_load_async_to_lds_b128 %0, %1, %2"
                 :: "v"(ldsoff), "v"(voff), "s"(sbase) : "memory");
  }
  asm volatile("s_wait_asynccnt 0x0" ::: "memory");

  // ---- Stage 2: per-row partial sums.  pq[h*16+n]: n<8 -> P, n>=8 -> Q ----
  float* pq = wl + PQ_OFF;
  for (int t = 0; t < 14; ++t) {          // 448 (h,bw) entries / 32 lanes
    int e  = lane + 32 * t;
    int h  = e >> 3;
    int bw = e & 7;
    const float* r = wl + h * HW + bw * BLK;
    float p = 0.f, q = 0.f;
#pragma unroll
    for (int i = 0; i < BLK; ++i) { float v = r[i]; p += v; q = fmaf(v, v, q); }
    pq[h * 16 + bw]     = p;
    pq[h * 16 + 8 + bw] = q;
  }
  asm volatile("s_wait_dscnt 0x0" ::: "memory");

  // ---- Stage 3: column reduction via V_WMMA_F32_16X16X4_F32 ----
  // D[m][n] = sum_h [h/7 == m] * pq[h][n]  (m = bh; n<8 -> sum, n>=8 -> sumsq)
  const int n  = lane & 15;          // M for A, N for B/D
  const int kh = (lane >> 4) << 1;   // K base of this lane half: 0 or 2
  v8f acc = {};
#pragma unroll
  for (int kc = 0; kc < 14; ++kc) {  // 56 rows = 14 chunks of K=4
    int r0 = 4 * kc + kh;
    int r1 = r0 + 1;
    v2f a, b;
    a.x = (r0 / 7 == n) ? 1.0f : 0.0f;   // block-membership indicator
    a.y = (r1 / 7 == n) ? 1.0f : 0.0f;
    b.x = pq[r0 * 16 + n];
    b.y = pq[r1 * 16 + n];
    acc = __builtin_amdgcn_wmma_f32_16x16x4_f32(false, a, false, b,
                                                (short)0, acc, false, false);
  }

  // ---- Stage 4: D rows live in lanes 0-15; stash, combine, store ----
  float* dmat = wl + D_OFF;
  if (lane < 16) {
#pragma unroll
    for (int j = 0; j < 8; ++j) dmat[j * 16 + n] = acc[j];  // D[M=j][N=lane]
  }
  asm volatile("s_wait_dscnt 0x0" ::: "memory");

  const float inv = 1.0f / 49.0f;
#pragma unroll
  for (int half = 0; half < 2; ++half) {
    int o  = lane + 32 * half;   // 0..63 output blocks of this slice
    int bh = o >> 3, bw = o & 7;
    float S    = dmat[bh * 16 + bw];
    float Qs   = dmat[bh * 16 + 8 + bw];
    float mean = S * inv;
    float msq  = Qs * inv;
    float var  = msq - mean * mean;
    // Faithful reference bug: last block row/col use mean(x^2) (zero mean).
    out[(size_t)s * 64 + o] = (bh == 7 || bw == 7) ? msq : var;
  }
}

extern "C" void kernel_launch(void* const* d_in, const int* in_sizes, int n_in,
                              void* d_out, int out_size, void* d_ws, size_t ws_size,
                              hipStream_t stream) {
  (void)in_sizes; (void)n_in; (void)d_ws; (void)ws_size; (void)out_size;
  const float* x = (const float*)d_in[0];   // fp32 (32,512,56,56)
  float* out = (float*)d_out;               // fp32 (32,512,8,8)
  dim3 grid(NSLICE / WAVES_PER_BLOCK);      // 8192 blocks of 64 threads (2 waves)
  approximator_blockmse_kernel<<<grid, WAVES_PER_BLOCK * 32, 0, stream>>>(x, out);
}